// Phase2ALayer_22402549416003
// MI455X (gfx1250) — compile-verified
//
#include <hip/hip_runtime.h>
#include <math.h>

typedef __attribute__((ext_vector_type(2))) float v2f;
typedef __attribute__((ext_vector_type(4))) float v4f;
typedef __attribute__((ext_vector_type(4))) int   v4i;
typedef __attribute__((ext_vector_type(8))) float v8f;

#define INV_TEMP 10.0f
#define SINK_ITERS 20

// ---------------------------------------------------------------------------
// Setup kernel: one wave32. Sinkhorn on the 4x8 log_alpha entirely in
// registers via shfl_xor butterfly logsumexp reductions, then the tiny
// (8x4) = R.T(8x4) @ logic_masks(4x4) product on the matrix core using
// V_WMMA_F32_16X16X4_F32 (zero-padded to 16x4 * 4x16 -> 16x16).
// Result temporal_masks (8,4) row-major -> d_ws.
// ---------------------------------------------------------------------------
__global__ __launch_bounds__(32)
void phase2a_setup(const float* __restrict__ logic_masks, // (4,4)
                   const float* __restrict__ log_alpha,   // (4,8)
                   const float* __restrict__ sign_logits, // (8,)
                   float* __restrict__ tm_out)            // (8,4)
{
    const int lane = threadIdx.x;        // lane = parent*8 + child
    const int child = lane & 7;

    float la = log_alpha[lane] * INV_TEMP;

    #pragma unroll 1
    for (int it = 0; it < SINK_ITERS; ++it) {
        // column logsumexp over parents (lanes differing in bits 3,4)
        float m = la;
        m = fmaxf(m, __shfl_xor(m, 8, 32));
        m = fmaxf(m, __shfl_xor(m, 16, 32));
        float e = __expf(la - m);
        e += __shfl_xor(e, 8, 32);
        e += __shfl_xor(e, 16, 32);
        la -= m + __logf(e);
        // row logsumexp over children (lanes differing in bits 0..2)
        float mr = la;
        mr = fmaxf(mr, __shfl_xor(mr, 1, 32));
        mr = fmaxf(mr, __shfl_xor(mr, 2, 32));
        mr = fmaxf(mr, __shfl_xor(mr, 4, 32));
        float er = __expf(la - mr);
        er += __shfl_xor(er, 1, 32);
        er += __shfl_xor(er, 2, 32);
        er += __shfl_xor(er, 4, 32);
        la += 0.6931471805599453f - (mr + __logf(er)); // + log(n_child/n_parent)
    }

    const float P   = __expf(la);
    const float sgn = tanhf(sign_logits[child] * INV_TEMP);
    const float R   = P * sgn;           // R[p][c] at lane p*8+c

    __shared__ float sR[32];             // R row-major (4,8)
    __shared__ float sL[16];             // logic_masks row-major (4,4)
    sR[lane] = R;
    if (lane < 16) sL[lane] = logic_masks[lane];
    __syncthreads();

    // A (16x4 f32): A[m][k] = R[k][m], m=child (zero pad m>=8), k=parent.
    // ISA layout: lanes 0-15 -> M=lane, VGPR0=K0/VGPR1=K1; lanes 16-31 -> K2/K3.
    const int half = lane >> 4;
    const int mrow = lane & 15;
    const int k0   = half * 2;
    v2f amat, bmat;
    amat.x = (mrow < 8) ? sR[(k0 + 0) * 8 + mrow] : 0.0f;
    amat.y = (mrow < 8) ? sR[(k0 + 1) * 8 + mrow] : 0.0f;
    // B (4x16 f32): B[k][n] = logic_masks[k][n], zero pad n>=4 (mirrored layout).
    bmat.x = (mrow < 4) ? sL[(k0 + 0) * 4 + mrow] : 0.0f;
    bmat.y = (mrow < 4) ? sL[(k0 + 1) * 4 + mrow] : 0.0f;

    v8f cmat = {};
    v8f d = __builtin_amdgcn_wmma_f32_16x16x4_f32(
        /*neg_a=*/false, amat, /*neg_b=*/false, bmat,
        /*c_mod=*/(short)0, cmat, /*reuse_a=*/false, /*reuse_b=*/false);

    // D: lane n (0-15) = column n; VGPR j = row M=j for lanes 0-15.
    if (lane < 4) {
        #pragma unroll
        for (int j = 0; j < 8; ++j)
            tm_out[j * 4 + lane] = d[j];   // temporal_masks[child=j][feat=lane]
    }
}

// ---------------------------------------------------------------------------
// Main kernel: pure HBM stream. 128-bit non-temporal loads of a/b/op_id,
// per-element 8-way LDS table lookup (ds_load_b128), 3 FMAs + tanh,
// 128-bit non-temporal store. 16 B/element -> HBM roofline ~11.5 us.
// ---------------------------------------------------------------------------
__global__ __launch_bounds__(256)
void phase2a_main(const float* __restrict__ a, const float* __restrict__ b,
                  const int* __restrict__ op, const float* __restrict__ tm,
                  float* __restrict__ out, int n4)
{
    __shared__ v4f s_tm[8];
    if (threadIdx.x < 8) {
        s_tm[threadIdx.x] = ((const v4f*)tm)[threadIdx.x];
    }
    __syncthreads();

    const int i = blockIdx.x * blockDim.x + threadIdx.x;
    if (i >= n4) return;

    v4f av = __builtin_nontemporal_load((const v4f*)a + i);
    v4f bv = __builtin_nontemporal_load((const v4f*)b + i);
    v4i iv = __builtin_nontemporal_load((const v4i*)op + i);

    v4f ov;
    #pragma unroll
    for (int j = 0; j < 4; ++j) {
        v4f m = s_tm[iv[j] & 7];
        float x = m.x + m.y * av[j] + m.z * bv[j] + m.w * (av[j] * bv[j]);
        ov[j] = tanhf(x * 10.0f);
    }
    __builtin_nontemporal_store(ov, (v4f*)out + i);
}

// ---------------------------------------------------------------------------
extern "C" void kernel_launch(void* const* d_in, const int* in_sizes, int n_in,
                              void* d_out, int out_size, void* d_ws, size_t ws_size,
                              hipStream_t stream) {
    const float* a  = (const float*)d_in[0];   // (B,)
    const float* b  = (const float*)d_in[1];   // (B,)
    const int*   op = (const int*)  d_in[2];   // (B,)
    const float* lm = (const float*)d_in[3];   // (4,4)
    const float* la = (const float*)d_in[4];   // (4,8)
    const float* sl = (const float*)d_in[5];   // (8,)
    float* out = (float*)d_out;
    float* tm  = (float*)d_ws;                 // 32 floats scratch

    phase2a_setup<<<1, 32, 0, stream>>>(lm, la, sl, tm);

    const int n  = in_sizes[0];
    const int n4 = n >> 2;                     // B = 2^24, divisible by 4
    const int block = 256;
    const int grid  = (n4 + block - 1) / block;
    phase2a_main<<<grid, block, 0, stream>>>(a, b, op, tm, out, n4);
}